// subspace_linop_48541720380100
// MI455X (gfx1250) — compile-verified
//
#include <hip/hip_runtime.h>
#include <math.h>

typedef float v2f __attribute__((ext_vector_type(2)));
typedef float v8f __attribute__((ext_vector_type(8)));
typedef unsigned int v4u __attribute__((ext_vector_type(4)));
typedef int v8i __attribute__((ext_vector_type(8)));
typedef int v4i __attribute__((ext_vector_type(4)));

#define NN     256
#define NSUBS  4
#define NCOILS 8
#define NPLANE 32          // NCOILS*NSUBS
#define NRO    1024
#define NPE    16
#define NTR    100
#define PLANE  65536       // NN*NN
#define CPLANE 131072      // re plane + im plane

// workspace layout (float offsets)
#define WS_F    0                          // 2*PLANE floats (F_re, F_im)
#define WS_IMG  (2*PLANE)                  // NPLANE*CPLANE floats
#define WS_T    (WS_IMG + NPLANE*CPLANE)   // NPLANE*CPLANE floats
#define WS_G    WS_IMG                     // reuse img region (dead after GEMM-1)

#if defined(__has_builtin)
#if __has_builtin(__builtin_amdgcn_tensor_load_to_lds)
#define HAVE_TDM 1
#endif
#endif
#ifndef HAVE_TDM
#define HAVE_TDM 0
#endif

#define WMMA_F32(A, B, C) \
    __builtin_amdgcn_wmma_f32_16x16x4_f32(false, A, false, B, (short)0, C, false, false)

// F[k,n] = (1/16) * (-1)^(k+n) * exp(-2*pi*i*k*n/256)  (symmetric; folds both
// fftshifts and the ortho norm), so centered-fft2 == F * img * F.
__global__ void build_F(float* __restrict__ F) {
    int idx = blockIdx.x * blockDim.x + threadIdx.x;
    if (idx >= PLANE) return;
    int k = idx >> 8, n = idx & 255;
    float ang = -6.283185307179586f * (float)((k * n) & 255) * (1.0f / 256.0f);
    float s, c;
    __sincosf(ang, &s, &c);
    float sg = ((k + n) & 1) ? -0.0625f : 0.0625f;
    F[idx]         = sg * c;   // Re
    F[PLANE + idx] = sg * s;   // Im
}

// img[c*4+s] = mps[c] * alphas[s], split into planar Re/Im for the GEMMs.
__global__ void build_img(const float* __restrict__ alphas,
                          const float* __restrict__ mps,
                          float* __restrict__ img) {
    int idx = blockIdx.x * blockDim.x + threadIdx.x;   // NPLANE*PLANE
    if (idx >= NPLANE * PLANE) return;
    int p   = idx >> 16;
    int pix = idx & (PLANE - 1);
    int c = p >> 2, s = p & 3;
    float2 a = ((const float2*)alphas)[s * PLANE + pix];
    float2 m = ((const float2*)mps)[c * PLANE + pix];
    img[p * CPLANE + pix]         = m.x * a.x - m.y * a.y;
    img[p * CPLANE + PLANE + pix] = m.x * a.y + m.y * a.x;
}

// ---- Tensor Data Mover: 2-D tile (global f32, row stride 256 elems) -> LDS.
// D# built per CDNA5 ISA 8.3/8.4: group0 {count=1, lds_addr, global_addr,
// type=2}; group1 {data_size=4B, tensor 256x256, tile WxH, stride0=256,
// optional 1-DWORD pad every 2^(pad_code+1) DWORDs (== one pad per tile row)}.
__device__ __forceinline__ void tdm_load_2d(unsigned lds_off, const float* g,
                                            unsigned tile_w, unsigned tile_h,
                                            int pad_code) {
#if HAVE_TDM
    unsigned long long ga = (unsigned long long)(uintptr_t)g;
    v4u g0;
    g0.x = 1u;                                   // count=1 (valid user D#)
    g0.y = lds_off;                              // lds_addr (bytes)
    g0.z = (unsigned)ga;                         // global_addr[31:0]
    g0.w = (unsigned)(ga >> 32) | (2u << 30);    // global_addr[56:32] | type=2
    v8i g1;
    unsigned f0 = (2u << 16);                    // data_size = 4 bytes
    if (pad_code >= 0)
        f0 |= (1u << 20) | ((unsigned)pad_code << 22);  // pad_amount=1 DWORD
    g1[0] = (int)f0;                             // wg_mask=0 (not in cluster)
    g1[1] = (int)(256u << 16);                   // tensor_dim0 = 256 (lo16)
    g1[2] = (int)(256u << 16);                   // dim0 hi=0 | tensor_dim1=256
    g1[3] = (int)(tile_w << 16);                 // dim1 hi=0 | tile_dim0
    g1[4] = (int)tile_h;                         // tile_dim1 | tile_dim2=0
    g1[5] = 256;                                 // tensor_dim0_stride lo32
    g1[6] = 0;                                   // stride0 hi | stride1 lo
    g1[7] = 0;
    v4i z4 = {0, 0, 0, 0};                       // 2-D tensor: groups 2/3 unused
    v8i z8 = {0, 0, 0, 0, 0, 0, 0, 0};
    __builtin_amdgcn_tensor_load_to_lds(g0, g1, z4, z4, z8, 0);
#endif
}

__device__ __forceinline__ void store_tile(const v8f& cr, const v8f& ci,
                                           int rbase, int col, int plane,
                                           float* outPlanar, float* outGather,
                                           int mode) {
    if (mode == 0) {
        float* outRe = outPlanar + (long)plane * CPLANE;
        #pragma unroll
        for (int j = 0; j < 8; ++j) {
            outRe[(rbase + j) * NN + col]         = cr[j];
            outRe[PLANE + (rbase + j) * NN + col] = ci[j];
        }
    } else {
        #pragma unroll
        for (int j = 0; j < 8; ++j) {
            float2 v; v.x = cr[j]; v.y = ci[j];
            ((float2*)outGather)[(long)((rbase + j) * NN + col) * NPLANE + plane] = v;
        }
    }
}

// Complex GEMM C = A*B (256x256x256 per plane, 32 planes) on fp32 WMMA.
// 256 threads = 8 waves per block; block computes a 64x64 complex tile.
// K is consumed in 32-wide slices, double-buffered in LDS by the TDM (padded
// pitch: A rows 33 floats, B rows 65 floats -> conflict-free ds_load reads).
// Waves 0-3 issue slice k+1 into the alternate buffer, then wait
// tensorcnt<=1: TDM ops complete in order per wave, so slice k is resident
// while slice k+1 streams in behind the WMMA work.
__global__ void __launch_bounds__(256) wmma_cgemm(
        const float* __restrict__ Abase, long aStride,
        const float* __restrict__ Bbase, long bStride,
        float* __restrict__ outPlanar, float* __restrict__ outGather, int mode) {
    __shared__ float Ars[2][64 * 33];
    __shared__ float Ais[2][64 * 33];
    __shared__ float Brs[2][32 * 65];
    __shared__ float Bis[2][32 * 65];

    const int lane  = threadIdx.x & 31;
    const int wave  = threadIdx.x >> 5;
    const int plane = blockIdx.x >> 4;              // 16 blocks / plane
    const int m0 = (((blockIdx.x >> 2) & 3)) * 64;  // block row origin
    const int n0 = (blockIdx.x & 3) * 64;           // block col origin

    const float* Ar = Abase + (long)plane * aStride;
    const float* Ai = Ar + PLANE;
    const float* Br = Bbase + (long)plane * bStride;
    const float* Bi = Br + PLANE;

    // wave -> 16-row strip (wave&3) x two 16-col subtiles at (wave>>2)*32
    const int mloc  = (wave & 3) * 16 + (lane & 15);   // A row in [0,64)
    const int khalf = (lane >> 4) << 1;                // K offset 0 or 2
    const int nloc  = (wave >> 2) * 32 + (lane & 15);  // first B col in [0,64)

#if HAVE_TDM
    auto issue_slice = [&](int buf, int kk) {
        if      (wave == 0) tdm_load_2d((unsigned)(uintptr_t)(void*)&Ars[buf][0],
                                        Ar + m0 * NN + kk, 32, 64, 4);
        else if (wave == 1) tdm_load_2d((unsigned)(uintptr_t)(void*)&Ais[buf][0],
                                        Ai + m0 * NN + kk, 32, 64, 4);
        else if (wave == 2) tdm_load_2d((unsigned)(uintptr_t)(void*)&Brs[buf][0],
                                        Br + kk * NN + n0, 64, 32, 5);
        else                tdm_load_2d((unsigned)(uintptr_t)(void*)&Bis[buf][0],
                                        Bi + kk * NN + n0, 64, 32, 5);
    };
#endif

    v8f cr0 = {}, ci0 = {}, cr1 = {}, ci1 = {};

#if HAVE_TDM
    if (wave < 4) issue_slice(0, 0);     // prime the pipeline
#endif

    for (int ks = 0; ks < 8; ++ks) {     // 8 K-slices of 32
        const int cur = ks & 1;
#if HAVE_TDM
        if (wave < 4) {
            if (ks + 1 < 8) {
                issue_slice(cur ^ 1, (ks + 1) * 32);   // prefetch next slice
                __builtin_amdgcn_s_wait_tensorcnt(1);  // slice ks is resident
            } else {
                __builtin_amdgcn_s_wait_tensorcnt(0);
            }
        }
        __asm__ volatile("" ::: "memory");   // TDM wrote LDS behind the compiler
#else
        const int k = ks * 32;
        for (int i = threadIdx.x; i < 64 * 32; i += 256) {
            int r = i >> 5, c = i & 31;
            Ars[cur][r * 33 + c] = Ar[(m0 + r) * NN + k + c];
            Ais[cur][r * 33 + c] = Ai[(m0 + r) * NN + k + c];
        }
        for (int i = threadIdx.x; i < 32 * 64; i += 256) {
            int r = i >> 6, c = i & 63;
            Brs[cur][r * 65 + c] = Br[(k + r) * NN + n0 + c];
            Bis[cur][r * 65 + c] = Bi[(k + r) * NN + n0 + c];
        }
#endif
        __syncthreads();

        const float* As  = &Ars[cur][0];
        const float* Ais_ = &Ais[cur][0];
        const float* Bs  = &Brs[cur][0];
        const float* Bis_ = &Bis[cur][0];

        #pragma unroll
        for (int kk = 0; kk < 32; kk += 4) {
            int k0 = kk + khalf;
            v2f ar, ai, br0, bi0, br1, bi1;
            ar.x  = As[mloc * 33 + k0];    ar.y  = As[mloc * 33 + k0 + 1];
            ai.x  = Ais_[mloc * 33 + k0];  ai.y  = Ais_[mloc * 33 + k0 + 1];
            br0.x = Bs[k0 * 65 + nloc];         br0.y = Bs[(k0 + 1) * 65 + nloc];
            bi0.x = Bis_[k0 * 65 + nloc];       bi0.y = Bis_[(k0 + 1) * 65 + nloc];
            br1.x = Bs[k0 * 65 + nloc + 16];    br1.y = Bs[(k0 + 1) * 65 + nloc + 16];
            bi1.x = Bis_[k0 * 65 + nloc + 16];  bi1.y = Bis_[(k0 + 1) * 65 + nloc + 16];
            v2f nai = -ai;                 // f32 WMMA has no A-neg modifier
            cr0 = WMMA_F32(ar,  br0, cr0);
            cr0 = WMMA_F32(nai, bi0, cr0);
            ci0 = WMMA_F32(ar,  bi0, ci0);
            ci0 = WMMA_F32(ai,  br0, ci0);
            cr1 = WMMA_F32(ar,  br1, cr1);
            cr1 = WMMA_F32(nai, bi1, cr1);
            ci1 = WMMA_F32(ar,  bi1, ci1);
            ci1 = WMMA_F32(ai,  br1, ci1);
        }
        __syncthreads();   // buffer (cur) free for the slice-(ks+2) DMA
    }

    // C/D layout: VGPR j -> row rbase+j, col = lane&15 (+8 rows for lanes>=16)
    const int rbase = m0 + (wave & 3) * 16 + ((lane >> 4) << 3);
    const int colg  = n0 + (wave >> 2) * 32 + (lane & 15);
    store_tile(cr0, ci0, rbase, colg,      plane, outPlanar, outGather, mode);
    store_tile(cr1, ci1, rbase, colg + 16, plane, outPlanar, outGather, mode);
}

// One thread per (r,p,t): gather the contiguous 32 (c,s) complex values at the
// rounded trajectory point (L2-resident 256B line) and contract with phi.
__global__ void gather_phi(const float* __restrict__ G,
                           const float* __restrict__ trj,
                           const float* __restrict__ phi,
                           float* __restrict__ out) {
    const int TOTAL = NRO * NPE * NTR;
    int idx = blockIdx.x * blockDim.x + threadIdx.x;
    if (idx >= TOTAL) return;
    int t = idx % NTR;
    int p = (idx / NTR) % NPE;
    int r = idx / (NTR * NPE);

    float2 tr = ((const float2*)trj)[idx];
    int ix = (__float2int_rn(tr.x) + 128) & 255;   // round-half-even == jnp.round
    int iy = (__float2int_rn(tr.y) + 128) & 255;

    const float2* g = (const float2*)G + (long)(ix * NN + iy) * NPLANE;

    float2 ph[NSUBS];
    #pragma unroll
    for (int s = 0; s < NSUBS; ++s) ph[s] = ((const float2*)phi)[s * NTR + t];

    #pragma unroll
    for (int c = 0; c < NCOILS; ++c) {
        float re = 0.f, im = 0.f;
        #pragma unroll
        for (int s = 0; s < NSUBS; ++s) {
            float2 gv = g[c * NSUBS + s];
            float2 pv = ph[s];
            re += gv.x * pv.x - gv.y * pv.y;
            im += gv.x * pv.y + gv.y * pv.x;
        }
        float2 o; o.x = re; o.y = im;
        ((float2*)out)[((long)(c * NRO + r) * NPE + p) * NTR + t] = o;
    }
}

extern "C" void kernel_launch(void* const* d_in, const int* in_sizes, int n_in,
                              void* d_out, int out_size, void* d_ws, size_t ws_size,
                              hipStream_t stream) {
    const float* alphas = (const float*)d_in[0];   // (4,256,256) complex interleaved
    const float* mps    = (const float*)d_in[1];   // (8,256,256) complex interleaved
    const float* phi    = (const float*)d_in[2];   // (4,100)     complex interleaved
    const float* trj    = (const float*)d_in[3];   // (1024,16,100,2) float32

    float* ws  = (float*)d_ws;
    float* F   = ws + WS_F;
    float* img = ws + WS_IMG;
    float* T   = ws + WS_T;
    float* G   = ws + WS_G;   // reuses img region (img dead after GEMM-1)
    float* out = (float*)d_out;

    build_F  <<<PLANE / 256, 256, 0, stream>>>(F);
    build_img<<<(NPLANE * PLANE) / 256, 256, 0, stream>>>(alphas, mps, img);
    // T = F * img   (A = F shared across planes)
    wmma_cgemm<<<512, 256, 0, stream>>>(F, 0L, img, (long)CPLANE, T, nullptr, 0);
    // G = T * F, written in [k1][k2][plane] gather layout
    wmma_cgemm<<<512, 256, 0, stream>>>(T, (long)CPLANE, F, 0L, nullptr, G, 1);

    int total = NRO * NPE * NTR;
    gather_phi<<<(total + 255) / 256, 256, 0, stream>>>(G, trj, phi, out);
}